// BiLSTM_CRF_4879082848533
// MI455X (gfx1250) — compile-verified
//
#include <hip/hip_runtime.h>
#include <math.h>

// ---------------------------------------------------------------------------
// BiLSTM-CRF forward for MI455X (gfx1250), fp32 WMMA (V_WMMA_F32_16X16X4_F32)
// ---------------------------------------------------------------------------

typedef float v2f __attribute__((ext_vector_type(2)));
typedef float v8f __attribute__((ext_vector_type(8)));

#define T_LEN 256
#define B_SZ 64
#define EMB_D 300
#define HHALF 256      // hidden per direction
#define G4H 1024       // 4*HHALF gate width
#define KTAG 12
#define START_TAG 10
#define STOP_TAG 11
#define NEGV (-10000.0f)
#define HSTR 260       // padded LDS row stride (floats) to dodge bank conflicts

__device__ __forceinline__ float sigm(float x) {
  return 1.0f / (1.0f + __expf(-x));
}

// fp32 WMMA wrapper: D = A(16x4) * B(4x16) + C
__device__ __forceinline__ v8f wmma_f32(v2f a, v2f b, v8f c) {
  return __builtin_amdgcn_wmma_f32_16x16x4_f32(
      /*neg_a=*/false, a, /*neg_b=*/false, b,
      /*c_mod=*/(short)0, c, /*reuse_a=*/false, /*reuse_b=*/false);
}

// ---------------------------------------------------------------------------
// Kernel 0: pack weights into WMMA-B-friendly layouts.
//   w_ih [1024,300] -> wih_p: float2 at index (k2*1024 + n) = {W[n,2k2], W[n,2k2+1]}
//   w_hh [1024,256] -> whh_p: float2 at index (k2*1024 + n)
//   w_out [12,512]  -> wout_p: float2 at index (k2*16 + n), n>=12 zero-padded
// ---------------------------------------------------------------------------
__global__ void pack_weights(const float* __restrict__ wihf, const float* __restrict__ whhf,
                             const float* __restrict__ wihb, const float* __restrict__ whhb,
                             const float* __restrict__ wout,
                             float* wihf_p, float* whhf_p,
                             float* wihb_p, float* whhb_p, float* wout_p) {
  const int S_IH = 150 * 1024;   // float2 elements
  const int S_HH = 128 * 1024;
  const int S_WO = 256 * 16;
  const int total = 2 * S_IH + 2 * S_HH + S_WO;
  for (int i = blockIdx.x * blockDim.x + threadIdx.x; i < total;
       i += gridDim.x * blockDim.x) {
    int idx = i;
    if (idx < S_IH) {
      int k2 = idx >> 10, n = idx & 1023;
      wihf_p[2 * idx]     = wihf[n * EMB_D + 2 * k2];
      wihf_p[2 * idx + 1] = wihf[n * EMB_D + 2 * k2 + 1];
      continue;
    }
    idx -= S_IH;
    if (idx < S_HH) {
      int k2 = idx >> 10, n = idx & 1023;
      whhf_p[2 * idx]     = whhf[n * HHALF + 2 * k2];
      whhf_p[2 * idx + 1] = whhf[n * HHALF + 2 * k2 + 1];
      continue;
    }
    idx -= S_HH;
    if (idx < S_IH) {
      int k2 = idx >> 10, n = idx & 1023;
      wihb_p[2 * idx]     = wihb[n * EMB_D + 2 * k2];
      wihb_p[2 * idx + 1] = wihb[n * EMB_D + 2 * k2 + 1];
      continue;
    }
    idx -= S_IH;
    if (idx < S_HH) {
      int k2 = idx >> 10, n = idx & 1023;
      whhb_p[2 * idx]     = whhb[n * HHALF + 2 * k2];
      whhb_p[2 * idx + 1] = whhb[n * HHALF + 2 * k2 + 1];
      continue;
    }
    idx -= S_HH;
    { // wout
      int k2 = idx >> 4, n = idx & 15;
      float a = 0.f, b = 0.f;
      if (n < KTAG) {
        a = wout[n * 512 + 2 * k2];
        b = wout[n * 512 + 2 * k2 + 1];
      }
      wout_p[2 * idx] = a;
      wout_p[2 * idx + 1] = b;
    }
  }
}

// ---------------------------------------------------------------------------
// Kernel 1: embedding gather + input projection GEMM (both directions).
//   zin[dir][m=t*64+b][n] = x[m,:] @ W_ih^T + (b_ih + b_hh),  M=16384,K=300,N=1024
// One wave computes a 16x128 output strip (8 n-tiles), K-loop of 75 WMMA tiles.
// ---------------------------------------------------------------------------
__global__ void input_gemm(const int* __restrict__ sent, const float* __restrict__ embed,
                           const float* __restrict__ wihf_p, const float* __restrict__ wihb_p,
                           const float* __restrict__ bihf, const float* __restrict__ bhhf,
                           const float* __restrict__ bihb, const float* __restrict__ bhhb,
                           float* __restrict__ zin_f, float* __restrict__ zin_b) {
  int gwave = (blockIdx.x * blockDim.x + threadIdx.x) >> 5; // 0..16383
  int lane = threadIdx.x & 31;
  int ln = lane & 15;
  int hi = lane >> 4;            // 0: K={0,1}, 1: K={2,3} within each 4-wide K tile
  int dir = gwave >> 13;
  int rem = gwave & 8191;
  int mtile = rem >> 3;          // 0..1023
  int ng = rem & 7;              // 8 n-tiles each

  const float2* wp = (const float2*)(dir ? wihb_p : wihf_p);
  const float* bi = dir ? bihb : bihf;
  const float* bh = dir ? bhhb : bhhf;
  float* zout = dir ? zin_b : zin_f;

  // A row for this lane (rows striped over lanes%16)
  int row = mtile * 16 + ln;     // flat m = t*64 + b
  int t = row >> 6, b = row & 63;
  int tok = sent[b * T_LEN + t];
  const float2* e2 = (const float2*)embed;
  long long ebase = (long long)tok * (EMB_D / 2);

  v8f acc[8];
  for (int j = 0; j < 8; ++j) {
    int n = (ng * 8 + j) * 16 + ln;
    float bv = bi[n] + bh[n];
    for (int r = 0; r < 8; ++r) acc[j][r] = bv;
  }

  for (int kt = 0; kt < 75; ++kt) {          // K = 300 = 75 * 4
    int k2 = kt * 2 + hi;                    // float2 K index
    float2 av = e2[ebase + k2];
    v2f a; a.x = av.x; a.y = av.y;
    for (int j = 0; j < 8; ++j) {
      int n = (ng * 8 + j) * 16 + ln;
      float2 bv = wp[(long long)k2 * 1024 + n];
      v2f bb; bb.x = bv.x; bb.y = bv.y;
      acc[j] = wmma_f32(a, bb, acc[j]);
    }
  }

  for (int j = 0; j < 8; ++j) {
    int n = (ng * 8 + j) * 16 + ln;
    for (int r = 0; r < 8; ++r) {
      int mm = mtile * 16 + hi * 8 + r;      // D layout: lanes 16-31 hold rows 8-15
      zout[(long long)mm * G4H + n] = acc[j][r];
    }
  }
}

// ---------------------------------------------------------------------------
// Kernel 2: persistent recurrent scan, one workgroup per direction.
//   Per step: z = h @ W_hh^T + zin[t]  (64x256 @ 256x1024), then LSTM cell.
// Each wave owns one 16-wide j-column (jt == wave) and ALL four m-tiles plus
// all four gate columns: 16 live accumulators. B fragments (W_hh) are loaded
// once per k-tile and reused across the 4 m-tiles; A fragments come from the
// LDS-resident double-buffered h. c stays in registers for all 256 steps and
// the LSTM cell runs directly on WMMA accumulators (each thread owns i/f/g/o
// of the same (b,j) elements by construction of the D-matrix layout).
// ---------------------------------------------------------------------------
__global__ void lstm_scan(const float* __restrict__ zin_f, const float* __restrict__ zin_b,
                          const float* __restrict__ whhf_p, const float* __restrict__ whhb_p,
                          const float* __restrict__ h0, const float* __restrict__ c0,
                          float* __restrict__ hf_all, float* __restrict__ hb_all) {
  extern __shared__ float sh[];              // 2 * 64 * HSTR floats
  int dir = blockIdx.x;
  const float* zin = dir ? zin_b : zin_f;
  const float2* wp = (const float2*)(dir ? whhb_p : whhf_p);
  float* h_all = dir ? hb_all : hf_all;

  int tid = threadIdx.x;                     // 512 threads = 16 waves
  int wave = tid >> 5, lane = tid & 31;
  int ln = lane & 15, hi = lane >> 4;
  int j = wave * 16 + ln;                    // this wave's j-column block

  // load h0 into buffer 0
  for (int e = 0; e < 32; ++e) {
    int flat = tid + e * 512;                // 0..16383
    int b = flat >> 8, jj = flat & 255;
    sh[b * HSTR + jj] = h0[dir * 16384 + flat];
  }
  // c registers: 4 m-tiles x 8 rows each (thread owns (b, j) consistently)
  float creg[4][8];
  for (int m = 0; m < 4; ++m)
    for (int r = 0; r < 8; ++r) {
      int b = m * 16 + hi * 8 + r;
      creg[m][r] = c0[dir * 16384 + b * HHALF + j];
    }
  __syncthreads();

  for (int s = 0; s < T_LEN; ++s) {
    int tt = dir ? (T_LEN - 1 - s) : s;      // backward dir scans reversed time
    const float2* sh2 = (const float2*)(sh + (s & 1) * (64 * HSTR));
    float* shn = sh + ((s + 1) & 1) * (64 * HSTR);
    const float* zrow = zin + (long long)tt * (B_SZ * G4H);

    // 16 accumulators: [m-tile][gate], C init = input projection (incl biases)
    v8f acc[4][4];
    for (int m = 0; m < 4; ++m)
      for (int g = 0; g < 4; ++g) {
        int n = g * HHALF + j;
        for (int r = 0; r < 8; ++r) {
          int b = m * 16 + hi * 8 + r;
          acc[m][g][r] = zrow[b * G4H + n];
        }
      }

    for (int kt = 0; kt < 64; ++kt) {        // K = 256 = 64 * 4
      int k2 = kt * 2 + hi;
      if (kt + 4 < 64)                       // gfx1250 global_prefetch path
        __builtin_prefetch(&wp[(long long)(k2 + 8) * 1024 + j], 0, 0);
      // B fragments: one load per gate, shared by all 4 m-tiles
      v2f bfr[4];
      for (int g = 0; g < 4; ++g) {
        float2 bv = wp[(long long)k2 * 1024 + g * HHALF + j];
        bfr[g].x = bv.x; bfr[g].y = bv.y;
      }
      // A fragments from LDS h, one per m-tile
      for (int m = 0; m < 4; ++m) {
        float2 av = sh2[(m * 16 + ln) * (HSTR / 2) + k2];
        v2f a; a.x = av.x; a.y = av.y;
        for (int g = 0; g < 4; ++g)
          acc[m][g] = wmma_f32(a, bfr[g], acc[m][g]);
      }
    }

    // LSTM cell directly on accumulators (gate order i,f,g,o)
    for (int m = 0; m < 4; ++m)
      for (int r = 0; r < 8; ++r) {
        int b = m * 16 + hi * 8 + r;
        float iv = acc[m][0][r], fv = acc[m][1][r];
        float gv = acc[m][2][r], ov = acc[m][3][r];
        float cn = sigm(fv) * creg[m][r] + sigm(iv) * tanhf(gv);
        float hn = sigm(ov) * tanhf(cn);
        creg[m][r] = cn;
        shn[b * HSTR + j] = hn;
        h_all[(long long)tt * 16384 + b * HHALF + j] = hn;
      }
    __syncthreads();                         // publish h_t before step t+1
  }
}

// ---------------------------------------------------------------------------
// Kernel 3: output projection. feats[b,t,k] = [hf|hb] @ w_out^T + b_out.
//   M=16384, K=512, N=12 (padded to one 16-wide tile).
// ---------------------------------------------------------------------------
__global__ void proj_gemm(const float* __restrict__ hf_all, const float* __restrict__ hb_all,
                          const float* __restrict__ wout_p, const float* __restrict__ bout,
                          float* __restrict__ feats) {
  int mtile = (blockIdx.x * blockDim.x + threadIdx.x) >> 5; // 0..1023
  int lane = threadIdx.x & 31;
  int ln = lane & 15, hi = lane >> 4;
  const float2* wp = (const float2*)wout_p;

  v8f acc;
  float bv = (ln < KTAG) ? bout[ln] : 0.0f;
  for (int r = 0; r < 8; ++r) acc[r] = bv;

  int row = mtile * 16 + ln;
  int t = row >> 6, b = row & 63;
  const float* ha = hf_all + (long long)t * 16384 + b * HHALF;
  const float* hbp = hb_all + (long long)t * 16384 + b * HHALF;

  for (int kt = 0; kt < 128; ++kt) {         // K = 512 = 128 * 4
    int kk = kt * 4 + hi * 2;
    v2f a;
    if (kk < HHALF) { a.x = ha[kk];          a.y = ha[kk + 1]; }
    else            { a.x = hbp[kk - HHALF]; a.y = hbp[kk - HHALF + 1]; }
    int k2 = kt * 2 + hi;
    float2 bb2 = wp[k2 * 16 + ln];
    v2f bb; bb.x = bb2.x; bb.y = bb2.y;
    acc = wmma_f32(a, bb, acc);
  }
  if (ln < KTAG) {
    for (int r = 0; r < 8; ++r) {
      int mm = mtile * 16 + hi * 8 + r;
      int t2 = mm >> 6, b2 = mm & 63;
      feats[b2 * (T_LEN * KTAG) + t2 * KTAG + ln] = acc[r];
    }
  }
}

// ---------------------------------------------------------------------------
// Kernel 4: Viterbi decode, one thread per batch element.
//   out[0..63] = score; out[64 + b*T + t] = path tag (as float).
// ---------------------------------------------------------------------------
__global__ void viterbi(const float* __restrict__ feats, const float* __restrict__ trans,
                        int* __restrict__ bp, float* __restrict__ out) {
  __shared__ float tr[KTAG * KTAG];
  int tid = threadIdx.x;
  for (int i = tid; i < KTAG * KTAG; i += blockDim.x) tr[i] = trans[i];
  __syncthreads();
  int b = tid;
  float fv[KTAG];
  for (int k = 0; k < KTAG; ++k) fv[k] = NEGV;
  fv[START_TAG] = 0.0f;
  const float* fb = feats + b * (T_LEN * KTAG);
  for (int t = 0; t < T_LEN; ++t) {
    float nf[KTAG];
    for (int n = 0; n < KTAG; ++n) {
      float best = fv[0] + tr[n * KTAG];
      int bi = 0;
      for (int p = 1; p < KTAG; ++p) {
        float v = fv[p] + tr[n * KTAG + p];
        if (v > best) { best = v; bi = p; }  // first-index tie-break
      }
      bp[t * (B_SZ * KTAG) + b * KTAG + n] = bi;
      nf[n] = best + fb[t * KTAG + n];
    }
    for (int n = 0; n < KTAG; ++n) fv[n] = nf[n];
  }
  float best = fv[0] + tr[STOP_TAG * KTAG];
  int bi = 0;
  for (int p = 1; p < KTAG; ++p) {
    float v = fv[p] + tr[STOP_TAG * KTAG + p];
    if (v > best) { best = v; bi = p; }
  }
  out[b] = best;
  int tag = bi;
  out[B_SZ + b * T_LEN + (T_LEN - 1)] = (float)tag;
  for (int t = T_LEN - 2; t >= 0; --t) {
    tag = bp[(t + 1) * (B_SZ * KTAG) + b * KTAG + tag];
    out[B_SZ + b * T_LEN + t] = (float)tag;
  }
}

// ---------------------------------------------------------------------------
extern "C" void kernel_launch(void* const* d_in, const int* in_sizes, int n_in,
                              void* d_out, int out_size, void* d_ws, size_t ws_size,
                              hipStream_t stream) {
  const int*   sent  = (const int*)d_in[0];
  const float* embed = (const float*)d_in[1];
  const float* wihf  = (const float*)d_in[2];
  const float* whhf  = (const float*)d_in[3];
  const float* bihf  = (const float*)d_in[4];
  const float* bhhf  = (const float*)d_in[5];
  const float* wihb  = (const float*)d_in[6];
  const float* whhb  = (const float*)d_in[7];
  const float* bihb  = (const float*)d_in[8];
  const float* bhhb  = (const float*)d_in[9];
  const float* wout  = (const float*)d_in[10];
  const float* bout  = (const float*)d_in[11];
  const float* trans = (const float*)d_in[12];
  const float* h0    = (const float*)d_in[13];
  const float* c0    = (const float*)d_in[14];

  float* ws = (float*)d_ws;
  size_t off = 0;
  float* zin_f  = ws + off; off += (size_t)16384 * 1024;   // 64 MB
  float* zin_b  = ws + off; off += (size_t)16384 * 1024;   // 64 MB
  float* hf     = ws + off; off += (size_t)256 * 16384;    // 16 MB
  float* hb     = ws + off; off += (size_t)256 * 16384;    // 16 MB
  float* wihf_p = ws + off; off += (size_t)150 * 1024 * 2;
  float* whhf_p = ws + off; off += (size_t)128 * 1024 * 2;
  float* wihb_p = ws + off; off += (size_t)150 * 1024 * 2;
  float* whhb_p = ws + off; off += (size_t)128 * 1024 * 2;
  float* wout_p = ws + off; off += (size_t)256 * 16 * 2;
  float* feats  = ws + off; off += (size_t)B_SZ * T_LEN * KTAG;
  int*   bpws   = (int*)(ws + off); off += (size_t)T_LEN * B_SZ * KTAG;

  pack_weights<<<1024, 256, 0, stream>>>(wihf, whhf, wihb, whhb, wout,
                                         wihf_p, whhf_p, wihb_p, whhb_p, wout_p);

  // 16384 waves (2 dirs x 1024 m-tiles x 8 n-groups), 8 waves/block
  input_gemm<<<2048, 256, 0, stream>>>(sent, embed, wihf_p, wihb_p,
                                       bihf, bhhf, bihb, bhhb, zin_f, zin_b);

  // one persistent workgroup per direction; 2 * 64 * HSTR * 4 bytes of LDS
  lstm_scan<<<2, 512, 2 * 64 * HSTR * sizeof(float), stream>>>(
      zin_f, zin_b, whhf_p, whhb_p, h0, c0, hf, hb);

  // 1024 m-tile waves, 8 waves/block
  proj_gemm<<<128, 256, 0, stream>>>(hf, hb, wout_p, bout, feats);

  viterbi<<<1, 64, 0, stream>>>(feats, trans, bpws, (float*)d_out);
}